// VMambaLayer_up_14224931685039
// MI455X (gfx1250) — compile-verified
//
#include <hip/hip_runtime.h>
#include <hip/hip_bf16.h>
#include <math.h>

// ---------------------------------------------------------------------------
// VMamba up-layer for MI455X (gfx1250, wave32, WMMA).
// All dense linear algebra runs through V_WMMA_F32_16X16X4_F32 (f32 in/out,
// exact numerics). Working set (~116MB f32) is L2-resident on MI455X (192MB),
// so GEMM operands stream from global/L2. Each wave computes a 16x64 tile
// (4 accumulators, A-fragment reuse, no EXEC churn in the K-loop). The
// selective scan is the serial bottleneck: one persistent 256-thread block
// per batch, 16 states per thread in registers, per-token B/C via LDS.
// ---------------------------------------------------------------------------

#define NB      2
#define DIMC    128
#define HH      48
#define WW      48
#define H2      96
#define W2      96
#define LSEQ    (H2 * W2)          // 9216
#define NTOK    (NB * LSEQ)        // 18432
#define DIN     256                // D_INNER
#define DST     16                 // D_STATE
#define DTR     8                  // DT_RANK
#define DCONV   4

typedef __attribute__((ext_vector_type(2))) float v2f;
typedef __attribute__((ext_vector_type(8))) float v8f;

__device__ __forceinline__ float silu_f(float v) {
    return v / (1.0f + __expf(-v));
}

__device__ __forceinline__ v8f wmma4(v2f a, v2f b, v8f c) {
    return __builtin_amdgcn_wmma_f32_16x16x4_f32(
        /*neg_a=*/false, a, /*neg_b=*/false, b,
        /*c_mod=*/(short)0, c, /*reuse_a=*/false, /*reuse_b=*/false);
}

// ---------------------------------------------------------------------------
// f32 WMMA GEMM: C[M,N] (+)= A[M,K] * B[K,N], row-major with strides.
// Block = 4 waves; each wave owns one 16x64 tile (4 N-subtiles sharing the
// A fragment). M must be a multiple of 64, K of 4. Ragged N: OOB columns
// load from a clamped (valid) address — a D column depends only on the same
// B column, so garbage columns are simply never stored. No EXEC manipulation
// inside the K-loop; EXEC is all-1s at every v_wmma (ISA requirement).
// ---------------------------------------------------------------------------
__global__ void gemm_wmma_f32(const float* __restrict__ A, int lda,
                              const float* __restrict__ B, int ldb,
                              float* __restrict__ C, int ldc,
                              int N, int K, int accum)
{
    const int lane = threadIdx.x & 31;
    const int wv   = threadIdx.x >> 5;         // wave id 0..3
    const int half = lane >> 4;                // 0 | 1
    const int l16  = lane & 15;
    const int tileN = blockIdx.x * 64;
    const int tileM = (blockIdx.y * 4 + wv) * 16;

    const int m = tileM + l16;
    const float* __restrict__ Arow = A + (size_t)m * lda;

    int  ncl[4];
    bool nok[4];
#pragma unroll
    for (int j = 0; j < 4; ++j) {
        const int n = tileN + j * 16 + l16;
        nok[j] = (n < N);
        ncl[j] = nok[j] ? n : (N - 1);         // clamped, always in-bounds
    }

    v8f acc[4] = {};
    for (int k = 0; k < K; k += 4) {
        const int ka = k + half * 2;           // this lane's K pair
        v2f a;
        a.x = Arow[ka];
        a.y = Arow[ka + 1];
        const float* __restrict__ B0 = B + (size_t)ka * ldb;
        const float* __restrict__ B1 = B0 + ldb;
        v2f b0, b1, b2, b3;
        b0.x = B0[ncl[0]]; b0.y = B1[ncl[0]];
        b1.x = B0[ncl[1]]; b1.y = B1[ncl[1]];
        b2.x = B0[ncl[2]]; b2.y = B1[ncl[2]];
        b3.x = B0[ncl[3]]; b3.y = B1[ncl[3]];
        acc[0] = wmma4(a, b0, acc[0]);
        acc[1] = wmma4(a, b1, acc[1]);
        acc[2] = wmma4(a, b2, acc[2]);
        acc[3] = wmma4(a, b3, acc[3]);
    }

#pragma unroll
    for (int j = 0; j < 4; ++j) {
        if (!nok[j]) continue;
        const int n = tileN + j * 16 + l16;
#pragma unroll
        for (int r = 0; r < 8; ++r) {
            const size_t idx = (size_t)(tileM + r + half * 8) * ldc + n;
            if (accum) C[idx] += acc[j][r];
            else       C[idx]  = acc[j][r];
        }
    }
}

// ---------------------------------------------------------------------------
// x (B, 128, 48, 48) channel-first  ->  xt (B*2304, 128) token-major
// ---------------------------------------------------------------------------
__global__ void transpose_in(const float* __restrict__ x, float* __restrict__ xt)
{
    int idx = blockIdx.x * blockDim.x + threadIdx.x;       // over NB*2304*128
    if (idx >= NB * HH * WW * DIMC) return;
    int c = idx & (DIMC - 1);
    int p = (idx >> 7) % (HH * WW);
    int b = idx / (DIMC * HH * WW);
    xt[idx] = x[((size_t)(b * DIMC + c)) * (HH * WW) + p];
}

// ---------------------------------------------------------------------------
// pixel-shuffle(2x) of expand output + LayerNorm(64) + concat skip -> x (.,128)
// one wave per output token; each thread owns 2 of the 64 LN channels.
// ---------------------------------------------------------------------------
__global__ void shuffle_ln_concat(const float* __restrict__ eo,
                                  const float* __restrict__ skip,
                                  const float* __restrict__ pe_g,
                                  const float* __restrict__ pe_b,
                                  float* __restrict__ xcur)
{
    const int token = blockIdx.x;                 // b*9216 + h2*96 + w2
    const int b  = token / LSEQ;
    const int hw = token - b * LSEQ;
    const int h2 = hw / W2, w2 = hw - h2 * W2;
    const int h = h2 >> 1, w = w2 >> 1;
    const int q = ((h2 & 1) << 1) | (w2 & 1);     // d3*2 + d4

    const float* src = eo + ((size_t)((b * HH + h) * WW + w)) * 256 + q * 64;
    const int t = threadIdx.x;                    // 0..31

    float v0 = src[t], v1 = src[t + 32];
    float s = v0 + v1;
#pragma unroll
    for (int o = 16; o > 0; o >>= 1) s += __shfl_xor(s, o, 32);
    const float mu = s * (1.0f / 64.0f);
    const float d0 = v0 - mu, d1 = v1 - mu;
    float ss = d0 * d0 + d1 * d1;
#pragma unroll
    for (int o = 16; o > 0; o >>= 1) ss += __shfl_xor(ss, o, 32);
    const float rstd = rsqrtf(ss * (1.0f / 64.0f) + 1e-5f);

    float* dst = xcur + (size_t)token * DIMC;
    dst[t]      = d0 * rstd * pe_g[t]      + pe_b[t];
    dst[t + 32] = d1 * rstd * pe_g[t + 32] + pe_b[t + 32];

    const float* sk = skip + (size_t)b * 64 * LSEQ + hw;
    dst[64 + t]      = sk[(size_t)t * LSEQ];
    dst[64 + t + 32] = sk[(size_t)(t + 32) * LSEQ];
}

// ---------------------------------------------------------------------------
// per-token LayerNorm over 128 channels; one wave per token, 4 ch per thread.
// ---------------------------------------------------------------------------
__global__ void layernorm128(const float* __restrict__ x,
                             const float* __restrict__ g,
                             const float* __restrict__ bta,
                             float* __restrict__ out)
{
    const int token = blockIdx.x;
    const int t = threadIdx.x;
    const float* src = x + (size_t)token * DIMC;
    float v[4];
#pragma unroll
    for (int j = 0; j < 4; ++j) v[j] = src[t + 32 * j];
    float s = v[0] + v[1] + v[2] + v[3];
#pragma unroll
    for (int o = 16; o > 0; o >>= 1) s += __shfl_xor(s, o, 32);
    const float mu = s * (1.0f / 128.0f);
    float ss = 0.0f;
#pragma unroll
    for (int j = 0; j < 4; ++j) { v[j] -= mu; ss += v[j] * v[j]; }
#pragma unroll
    for (int o = 16; o > 0; o >>= 1) ss += __shfl_xor(ss, o, 32);
    const float rstd = rsqrtf(ss * (1.0f / 128.0f) + 1e-5f);
    float* dst = out + (size_t)token * DIMC;
#pragma unroll
    for (int j = 0; j < 4; ++j)
        dst[t + 32 * j] = v[j] * rstd * g[t + 32 * j] + bta[t + 32 * j];
}

// ---------------------------------------------------------------------------
// causal depthwise conv (kernel 4, per-batch padding) + SiLU.
// xz (., 512): channels [0,256) are xp.  -> xc (., 256)
// ---------------------------------------------------------------------------
__global__ void conv_silu(const float* __restrict__ xz,
                          const float* __restrict__ cw,   // (256,4)
                          const float* __restrict__ cb,   // (256)
                          float* __restrict__ xc)
{
    int idx = blockIdx.x * blockDim.x + threadIdx.x;      // over NTOK*256
    if (idx >= NTOK * DIN) return;
    const int d = idx & (DIN - 1);
    const int tok = idx >> 8;
    const int b = tok / LSEQ;
    const int l = tok - b * LSEQ;

    float acc = cb[d];
#pragma unroll
    for (int k = 0; k < DCONV; ++k) {
        const int lp = l + k - (DCONV - 1);
        if (lp >= 0)
            acc += cw[d * DCONV + k] *
                   xz[((size_t)(b * LSEQ + lp)) * 512 + d];
    }
    xc[idx] = silu_f(acc);
}

// ---------------------------------------------------------------------------
// dt = softplus(dt_pre + dt_bias)
// ---------------------------------------------------------------------------
__global__ void dt_softplus(float* __restrict__ dt, const float* __restrict__ dtb)
{
    int idx = blockIdx.x * blockDim.x + threadIdx.x;
    if (idx >= NTOK * DIN) return;
    const float v = dt[idx] + dtb[idx & (DIN - 1)];
    dt[idx] = (v > 20.0f) ? v : log1pf(__expf(v));
}

// ---------------------------------------------------------------------------
// selective scan: one block per batch, thread d owns channel d (16 states in
// registers); per-token B/C (32 floats) staged through LDS and broadcast.
// ---------------------------------------------------------------------------
__global__ void mamba_scan(const float* __restrict__ dt,
                           const float* __restrict__ xc,
                           const float* __restrict__ xdbl,  // (., 40)
                           const float* __restrict__ A_log, // (256,16)
                           float* __restrict__ y)
{
    const int b = blockIdx.x;
    const int d = threadIdx.x;     // 0..255
    float Ad[DST], h[DST];
#pragma unroll
    for (int s = 0; s < DST; ++s) {
        Ad[s] = -__expf(A_log[d * DST + s]);
        h[s] = 0.0f;
    }
    __shared__ float BC[2 * DST];  // [0,16)=B_t, [16,32)=C_t
    const size_t base = (size_t)b * LSEQ;
    for (int l = 0; l < LSEQ; ++l) {
        const size_t tok = base + l;
        if (d < 2 * DST) BC[d] = xdbl[tok * 40 + DTR + d];
        __syncthreads();
        const float dtv = dt[tok * DIN + d];
        const float xv  = xc[tok * DIN + d];
        const float dx  = dtv * xv;
        float acc = 0.0f;
#pragma unroll
        for (int s = 0; s < DST; ++s) {
            h[s] = h[s] * __expf(dtv * Ad[s]) + dx * BC[s];
            acc += h[s] * BC[DST + s];
        }
        y[tok * DIN + d] = acc;
        __syncthreads();
    }
}

// ---------------------------------------------------------------------------
// y = (y + xc*D) * silu(z),   z = xz[., 256+d]
// ---------------------------------------------------------------------------
__global__ void gate_mul(float* __restrict__ y,
                         const float* __restrict__ xc,
                         const float* __restrict__ xz,
                         const float* __restrict__ Dp)
{
    int idx = blockIdx.x * blockDim.x + threadIdx.x;
    if (idx >= NTOK * DIN) return;
    const int d = idx & (DIN - 1);
    const size_t tok = (size_t)(idx >> 8);
    const float z = xz[tok * 512 + DIN + d];
    y[idx] = (y[idx] + xc[idx] * Dp[d]) * silu_f(z);
}

// ---------------------------------------------------------------------------
// final bias add in-place on d_out (layout is a raw reshape: (B, L, 64) flat)
// ---------------------------------------------------------------------------
__global__ void bias_add64(float* __restrict__ out, const float* __restrict__ bias)
{
    int idx = blockIdx.x * blockDim.x + threadIdx.x;
    if (idx >= NTOK * 64) return;
    out[idx] += bias[idx & 63];
}

// ---------------------------------------------------------------------------
// host-side orchestration
// ---------------------------------------------------------------------------
extern "C" void kernel_launch(void* const* d_in, const int* in_sizes, int n_in,
                              void* d_out, int out_size, void* d_ws, size_t ws_size,
                              hipStream_t stream)
{
    (void)in_sizes; (void)n_in; (void)out_size; (void)ws_size;

    const float* x         = (const float*)d_in[0];
    const float* skip      = (const float*)d_in[1];
    const float* expand_w  = (const float*)d_in[2];
    const float* pe_g      = (const float*)d_in[3];
    const float* pe_b      = (const float*)d_in[4];
    const float* ln_g      = (const float*)d_in[5];
    const float* ln_b      = (const float*)d_in[6];
    const float* in_proj_w = (const float*)d_in[7];
    const float* conv_w    = (const float*)d_in[8];
    const float* conv_b    = (const float*)d_in[9];
    const float* x_proj_w  = (const float*)d_in[10];
    const float* dt_proj_w = (const float*)d_in[11];
    const float* dt_proj_b = (const float*)d_in[12];
    const float* A_log     = (const float*)d_in[13];
    const float* Dp        = (const float*)d_in[14];
    const float* out_proj_w= (const float*)d_in[15];
    const float* cbd_w     = (const float*)d_in[16];
    const float* cbd_b     = (const float*)d_in[17];

    float* ws = (float*)d_ws;
    // workspace layout (floats); total ~29.0M floats (~116 MB, L2-resident)
    float* xcur = ws;                                   // NTOK*128
    float* xln  = xcur + (size_t)NTOK * DIMC;           // NTOK*128 (also xt)
    float* xz   = xln  + (size_t)NTOK * DIMC;           // NTOK*512 (also expand out)
    float* xc   = xz   + (size_t)NTOK * 512;            // NTOK*256
    float* xdbl = xc   + (size_t)NTOK * DIN;            // NTOK*40
    float* dtb  = xdbl + (size_t)NTOK * 40;             // NTOK*256
    float* yb   = dtb  + (size_t)NTOK * DIN;            // NTOK*256

    const dim3 gblk(128);   // 4 waves; each wave = one 16x64 C tile

    // --- stage 1: channel-last transpose, expand GEMM, shuffle+LN+concat ---
    {
        int n = NB * HH * WW * DIMC;
        transpose_in<<<(n + 255) / 256, 256, 0, stream>>>(x, xln);
        // (4608 x 128) @ (128 x 256) -> expand out (reuse xz region)
        gemm_wmma_f32<<<dim3(256 / 64, (NB * HH * WW) / 64), gblk, 0, stream>>>(
            xln, DIMC, expand_w, 256, xz, 256, 256, DIMC, 0);
        shuffle_ln_concat<<<NTOK, 32, 0, stream>>>(xz, skip, pe_g, pe_b, xcur);
    }

    // --- stage 2: two Mamba mixer blocks ---
    for (int i = 0; i < 2; ++i) {
        layernorm128<<<NTOK, 32, 0, stream>>>(
            xcur, ln_g + i * DIMC, ln_b + i * DIMC, xln);

        // in_proj: (18432 x 128) @ (128 x 512)
        gemm_wmma_f32<<<dim3(512 / 64, NTOK / 64), gblk, 0, stream>>>(
            xln, DIMC, in_proj_w + (size_t)i * DIMC * 512, 512, xz, 512,
            512, DIMC, 0);

        {
            int n = NTOK * DIN;
            conv_silu<<<(n + 255) / 256, 256, 0, stream>>>(
                xz, conv_w + (size_t)i * DIN * DCONV, conv_b + (size_t)i * DIN, xc);
        }

        // x_proj: (18432 x 256) @ (256 x 40), ragged N
        gemm_wmma_f32<<<dim3(1, NTOK / 64), gblk, 0, stream>>>(
            xc, DIN, x_proj_w + (size_t)i * DIN * 40, 40, xdbl, 40,
            40, DIN, 0);

        // dt_proj: (18432 x 8) @ (8 x 256); A = first 8 cols of xdbl (lda=40)
        gemm_wmma_f32<<<dim3(DIN / 64, NTOK / 64), gblk, 0, stream>>>(
            xdbl, 40, dt_proj_w + (size_t)i * DTR * DIN, DIN, dtb, DIN,
            DIN, DTR, 0);
        {
            int n = NTOK * DIN;
            dt_softplus<<<(n + 255) / 256, 256, 0, stream>>>(
                dtb, dt_proj_b + (size_t)i * DIN);
        }

        // serial selective scan: 2 persistent blocks (one per batch)
        mamba_scan<<<NB, DIN, 0, stream>>>(
            dtb, xc, xdbl, A_log + (size_t)i * DIN * DST, yb);

        {
            int n = NTOK * DIN;
            gate_mul<<<(n + 255) / 256, 256, 0, stream>>>(
                yb, xc, xz, Dp + (size_t)i * DIN);
        }

        // out_proj with fused residual: xcur += (18432 x 256) @ (256 x 128)
        gemm_wmma_f32<<<dim3(DIMC / 64, NTOK / 64), gblk, 0, stream>>>(
            yb, DIN, out_proj_w + (size_t)i * DIN * DIMC, DIMC, xcur, DIMC,
            DIMC, DIN, 1);
    }

    // --- stage 3: cbd GEMM straight into d_out (raw reshape) + bias ---
    float* outp = (float*)d_out;
    gemm_wmma_f32<<<dim3(1, NTOK / 64), gblk, 0, stream>>>(
        xcur, DIMC, cbd_w, 64, outp, 64, 64, DIMC, 0);
    {
        int n = NTOK * 64;
        bias_add64<<<(n + 255) / 256, 256, 0, stream>>>(outp, cbd_b);
    }
}